// KpGNN_82695300317342
// MI455X (gfx1250) — compile-verified
//
#include <hip/hip_runtime.h>
#include <math.h>

typedef __attribute__((ext_vector_type(2))) float v2f;
typedef __attribute__((ext_vector_type(8))) float v8f;

// ---------- ordered-uint encoding for float atomic max ----------
#define ORD_NEG_INF 0x007FFFFFu  // f2ord(-inf)
__device__ __forceinline__ unsigned f2ord(float f) {
    unsigned b = __float_as_uint(f);
    return (b & 0x80000000u) ? ~b : (b | 0x80000000u);
}
__device__ __forceinline__ float ord2f(unsigned u) {
    return (u & 0x80000000u) ? __uint_as_float(u & 0x7FFFFFFFu)
                             : __uint_as_float(~u);
}

// ---------- init kernels ----------
__global__ void k_init_graph(float* cnt_pb, float* easum_pb,
                             float* cnt_sp, float* easum_sp,
                             float* out, int N) {
    int i = blockIdx.x * blockDim.x + threadIdx.x;
    if (i == 0) out[0] = 0.f;
    if (i < N) { cnt_pb[i] = 0.f; easum_pb[i] = 0.f; cnt_sp[i] = 0.f; easum_sp[i] = 0.f; }
}

__global__ void k_edge_deg(const int* __restrict__ ei, const float* __restrict__ ea,
                           float* cnt, float* easum, int E) {
    int i = blockIdx.x * blockDim.x + threadIdx.x;
    if (i >= E) return;
    int dst = ei[E + i];
    atomicAdd(&cnt[dst], 1.0f);
    atomicAdd(&easum[dst], ea[i]);
}

__global__ void k_loop_attr(const float* cnt, const float* easum, float* loopa, int N) {
    int i = blockIdx.x * blockDim.x + threadIdx.x;
    if (i < N) loopa[i] = easum[i] / fmaxf(cnt[i], 1.0f);
}

__global__ void k_layer_init(float* acc, float* denom, unsigned* mmax,
                             int N, int D, int H) {
    int i = blockIdx.x * blockDim.x + threadIdx.x;
    if (i < N * D) acc[i] = 0.f;
    if (i < N * H) { denom[i] = 0.f; mmax[i] = ORD_NEG_INF; }
}

// ---------- pack weights into zero-padded [Np x Kpad] (B-matrix layout) ----------
__global__ void k_pack_w(const float* __restrict__ W, float* __restrict__ Wp,
                         int Dout, int Din, int Kpad, int Np) {
    int i = blockIdx.x * blockDim.x + threadIdx.x;
    if (i >= Np * Kpad) return;
    int n = i / Kpad, k = i - n * Kpad;
    Wp[i] = (n < Dout && k < Din) ? W[n * Din + k] : 0.f;
}

// ---------- WMMA fp32 projection: xl = X*Wl^T + bl, xr = X*Wr^T + br ----------
// Xp: [N x Kpad] zero-padded row-major; WlP/WrP: [Np x Kpad] zero-padded.
// One wave = 16 nodes. Inner loop: 3x b64 loads + 2x V_WMMA_F32_16X16X4_F32.
__global__ void k_proj_wmma(const float* __restrict__ Xp, int Kpad,
                            const float* __restrict__ WlP, const float* __restrict__ bl,
                            const float* __restrict__ WrP, const float* __restrict__ br,
                            float* __restrict__ xl, float* __restrict__ xr,
                            int Dout, int Ntiles) {
    int wave = (blockIdx.x * blockDim.x + threadIdx.x) >> 5;
    int lane = threadIdx.x & 31;
    if (wave >= Ntiles) return;            // uniform per-wave exit: EXEC all ones inside
    const int m0   = wave * 16;
    const int half = lane >> 4;            // 0: K={0,1}, 1: K={2,3}
    const int lr   = lane & 15;
    const int kch  = Kpad >> 2;
    const int NT   = (Dout + 15) >> 4;

    const float* arow = Xp + (size_t)(m0 + lr) * Kpad + half * 2;

    for (int nt = 0; nt < NT; ++nt) {
        const int n = nt * 16 + lr;        // output column this lane owns (padded range)
        const float* bLrow = WlP + (size_t)n * Kpad + half * 2;
        const float* bRrow = WrP + (size_t)n * Kpad + half * 2;
        v8f accL = {0.f,0.f,0.f,0.f,0.f,0.f,0.f,0.f};
        v8f accR = {0.f,0.f,0.f,0.f,0.f,0.f,0.f,0.f};
        for (int kk = 0; kk < kch; ++kk) {
            v2f a  = *(const v2f*)(arow  + kk * 4);
            v2f bL = *(const v2f*)(bLrow + kk * 4);
            v2f bR = *(const v2f*)(bRrow + kk * 4);
            accL = __builtin_amdgcn_wmma_f32_16x16x4_f32(false, a, false, bL, (short)0, accL, false, false);
            accR = __builtin_amdgcn_wmma_f32_16x16x4_f32(false, a, false, bR, (short)0, accR, false, false);
        }
        if (n < Dout) {
            const float bLn = bl[n], bRn = br[n];
#pragma unroll
            for (int v = 0; v < 8; ++v) {
                const int mrow = m0 + v + 8 * half;
                xl[(size_t)mrow * Dout + n] = accL[v] + bLn;
                xr[(size_t)mrow * Dout + n] = accR[v] + bRn;
            }
        }
    }
}

// ---------- edge pass 1: segment max of logits ----------
template <int H, int C>
__global__ void k_logit_max(const int* __restrict__ ei, const float* __restrict__ ea,
                            const float* __restrict__ loopa,
                            const float* __restrict__ xl, const float* __restrict__ xr,
                            const float* __restrict__ We, const float* __restrict__ att,
                            unsigned* mmax, int E, int N) {
    constexpr int D = H * C;
    int i = blockIdx.x * blockDim.x + threadIdx.x;
    if (i >= E + N) return;
    int src, dst; float a;
    if (i < E) { src = ei[i]; dst = ei[E + i]; a = ea[i]; }
    else       { src = dst = i - E; a = loopa[src]; }

    float logit[H];
#pragma unroll
    for (int h = 0; h < H; ++h) logit[h] = 0.f;
#pragma unroll
    for (int j = 0; j < D; ++j) {
        float s = xl[(size_t)src * D + j] + xr[(size_t)dst * D + j] + a * We[j];
        s = (s > 0.f) ? s : 0.2f * s;
        logit[j / C] += s * att[j];
    }
#pragma unroll
    for (int h = 0; h < H; ++h)
        atomicMax(&mmax[dst * H + h], f2ord(logit[h]));
}

// ---------- edge pass 2: denom += ex ; acc += ex * xl[src] ----------
template <int H, int C>
__global__ void k_expsum(const int* __restrict__ ei, const float* __restrict__ ea,
                         const float* __restrict__ loopa,
                         const float* __restrict__ xl, const float* __restrict__ xr,
                         const float* __restrict__ We, const float* __restrict__ att,
                         const unsigned* __restrict__ mmax,
                         float* denom, float* acc, int E, int N) {
    constexpr int D = H * C;
    int i = blockIdx.x * blockDim.x + threadIdx.x;
    if (i >= E + N) return;
    int src, dst; float a;
    if (i < E) { src = ei[i]; dst = ei[E + i]; a = ea[i]; }
    else       { src = dst = i - E; a = loopa[src]; }

    float sl[D], logit[H];
#pragma unroll
    for (int h = 0; h < H; ++h) logit[h] = 0.f;
#pragma unroll
    for (int j = 0; j < D; ++j) {
        float v = xl[(size_t)src * D + j];
        sl[j] = v;
        float s = v + xr[(size_t)dst * D + j] + a * We[j];
        s = (s > 0.f) ? s : 0.2f * s;
        logit[j / C] += s * att[j];
    }
#pragma unroll
    for (int h = 0; h < H; ++h) {
        float ex = expf(logit[h] - ord2f(mmax[dst * H + h]));
        atomicAdd(&denom[dst * H + h], ex);
#pragma unroll
        for (int c = 0; c < C; ++c)
            atomicAdd(&acc[(size_t)dst * D + h * C + c], ex * sl[h * C + c]);
    }
}

// ---------- finalize: outPad[n*Kpad+j] = (j<D) ? relu(acc/denom + bias) : 0 ----------
__global__ void k_finalize(const float* __restrict__ acc, const float* __restrict__ denom,
                           const float* __restrict__ bias, float* __restrict__ outPad,
                           int Kpad, int N, int H, int C) {
    int i = blockIdx.x * blockDim.x + threadIdx.x;
    int D = H * C;
    if (i >= N * Kpad) return;
    int n = i / Kpad, j = i - n * Kpad;
    float v = 0.f;
    if (j < D) {
        v = acc[(size_t)n * D + j] / denom[n * H + j / C] + bias[j];
        v = (v > 0.f) ? v : 0.f;
    }
    outPad[i] = v;
}

// ---------- readout: per-edge 51->25->10->1 MLP, sum everything ----------
// h3p is zero-padded [N x 28]; first 25 entries per row are valid.
__global__ void k_edge_mlp(const int* __restrict__ ei, const float* __restrict__ d,
                           const float* __restrict__ h3p,
                           const float* __restrict__ W1, const float* __restrict__ b1,
                           const float* __restrict__ W2, const float* __restrict__ b2,
                           const float* __restrict__ W3, const float* __restrict__ b3,
                           float* out, int E) {
    __shared__ float red[256];
    int i = blockIdx.x * blockDim.x + threadIdx.x;
    float val = 0.f;
    if (i < E) {
        int src = ei[i], dst = ei[E + i];
        float emb[51];
#pragma unroll
        for (int j = 0; j < 25; ++j) emb[j]      = h3p[(size_t)src * 28 + j];
#pragma unroll
        for (int j = 0; j < 25; ++j) emb[25 + j] = h3p[(size_t)dst * 28 + j];
        emb[50] = d[i];
        float t1[25];
#pragma unroll
        for (int o = 0; o < 25; ++o) {
            float s = b1[o];
#pragma unroll
            for (int k = 0; k < 51; ++k) s += emb[k] * W1[o * 51 + k];
            t1[o] = (s > 0.f) ? s : 0.f;
        }
        float t2[10];
#pragma unroll
        for (int o = 0; o < 10; ++o) {
            float s = b2[o];
#pragma unroll
            for (int k = 0; k < 25; ++k) s += t1[k] * W2[o * 25 + k];
            t2[o] = (s > 0.f) ? s : 0.f;
        }
        float s = b3[0];
#pragma unroll
        for (int k = 0; k < 10; ++k) s += t2[k] * W3[k];
        val = s;
    }
    red[threadIdx.x] = val;
    __syncthreads();
    for (int off = blockDim.x / 2; off > 0; off >>= 1) {
        if ((int)threadIdx.x < off) red[threadIdx.x] += red[threadIdx.x + off];
        __syncthreads();
    }
    if (threadIdx.x == 0) atomicAdd(out, red[0]);
}

// ---------- host side ----------
static inline size_t al256(size_t x) { return (x + 255) & ~(size_t)255; }
static inline unsigned gblk(long n) { return (unsigned)((n + 255) / 256); }

extern "C" void kernel_launch(void* const* d_in, const int* in_sizes, int n_in,
                              void* d_out, int out_size, void* d_ws, size_t ws_size,
                              hipStream_t stream) {
    const float* x      = (const float*)d_in[0];
    const int*   pb_ei  = (const int*)d_in[1];
    const float* pb_ea  = (const float*)d_in[2];
    const int*   sp_ei  = (const int*)d_in[3];
    const float* sp_ea  = (const float*)d_in[4];
    const int*   if_ei  = (const int*)d_in[5];
    const float* if_ea  = (const float*)d_in[6];
    const float* c1_Wl = (const float*)d_in[7];  const float* c1_bl = (const float*)d_in[8];
    const float* c1_Wr = (const float*)d_in[9];  const float* c1_br = (const float*)d_in[10];
    const float* c1_We = (const float*)d_in[11]; const float* c1_at = (const float*)d_in[12];
    const float* c1_bs = (const float*)d_in[13];
    const float* c2_Wl = (const float*)d_in[14]; const float* c2_bl = (const float*)d_in[15];
    const float* c2_Wr = (const float*)d_in[16]; const float* c2_br = (const float*)d_in[17];
    const float* c2_We = (const float*)d_in[18]; const float* c2_at = (const float*)d_in[19];
    const float* c2_bs = (const float*)d_in[20];
    const float* c3_Wl = (const float*)d_in[21]; const float* c3_bl = (const float*)d_in[22];
    const float* c3_Wr = (const float*)d_in[23]; const float* c3_br = (const float*)d_in[24];
    const float* c3_We = (const float*)d_in[25]; const float* c3_at = (const float*)d_in[26];
    const float* c3_bs = (const float*)d_in[27];
    const float* e1_W = (const float*)d_in[28]; const float* e1_b = (const float*)d_in[29];
    const float* e2_W = (const float*)d_in[30]; const float* e2_b = (const float*)d_in[31];
    const float* e3_W = (const float*)d_in[32]; const float* e3_b = (const float*)d_in[33];

    const int N  = in_sizes[0] / 128;
    const int E1 = in_sizes[1] / 2;
    const int E2 = in_sizes[3] / 2;
    const int E3 = in_sizes[5] / 2;
    float* out = (float*)d_out;

    // workspace carve
    char* w = (char*)d_ws;
    size_t off = 0;
#define CARVE(ptr, ty, cnt) ty* ptr = (ty*)(w + off); off = al256(off + (size_t)(cnt) * sizeof(ty));
    CARVE(xl,    float,    (size_t)N * 25)
    CARVE(xr,    float,    (size_t)N * 25)
    CARVE(acc,   float,    (size_t)N * 25)
    CARVE(denom, float,    (size_t)N * 5)
    CARVE(mmax,  unsigned, (size_t)N * 5)
    CARVE(cnt_pb, float, N)  CARVE(easum_pb, float, N)  CARVE(loop_pb, float, N)
    CARVE(cnt_sp, float, N)  CARVE(easum_sp, float, N)  CARVE(loop_sp, float, N)
    CARVE(h1p, float, (size_t)N * 12)   // layer-1 output, Kpad=12 (9 used)
    CARVE(h2p, float, (size_t)N * 16)   // layer-2 output, Kpad=16 (15 used)
    CARVE(h3p, float, (size_t)N * 28)   // layer-3 output, Kpad=28 (25 used)
    CARVE(w1lp, float, 16 * 128) CARVE(w1rp, float, 16 * 128)  // layer-1 packed weights
    CARVE(w2lp, float, 16 * 12)  CARVE(w2rp, float, 16 * 12)   // layer-2
    CARVE(w3lp, float, 32 * 16)  CARVE(w3rp, float, 32 * 16)   // layer-3
#undef CARVE

    const int Ntiles = (N + 15) / 16;
    const unsigned projBlocks = gblk((long)Ntiles * 32);

    // ---- graph-level prep ----
    k_init_graph<<<gblk(N), 256, 0, stream>>>(cnt_pb, easum_pb, cnt_sp, easum_sp, out, N);
    k_edge_deg<<<gblk(E1), 256, 0, stream>>>(pb_ei, pb_ea, cnt_pb, easum_pb, E1);
    k_edge_deg<<<gblk(E2), 256, 0, stream>>>(sp_ei, sp_ea, cnt_sp, easum_sp, E2);
    k_loop_attr<<<gblk(N), 256, 0, stream>>>(cnt_pb, easum_pb, loop_pb, N);
    k_loop_attr<<<gblk(N), 256, 0, stream>>>(cnt_sp, easum_sp, loop_sp, N);

    // ---- pack weights (tiny) ----
    k_pack_w<<<gblk(16 * 128), 256, 0, stream>>>(c1_Wl, w1lp, 9, 128, 128, 16);
    k_pack_w<<<gblk(16 * 128), 256, 0, stream>>>(c1_Wr, w1rp, 9, 128, 128, 16);
    k_pack_w<<<gblk(16 * 12), 256, 0, stream>>>(c2_Wl, w2lp, 15, 9, 12, 16);
    k_pack_w<<<gblk(16 * 12), 256, 0, stream>>>(c2_Wr, w2rp, 15, 9, 12, 16);
    k_pack_w<<<gblk(32 * 16), 256, 0, stream>>>(c3_Wl, w3lp, 25, 15, 16, 32);
    k_pack_w<<<gblk(32 * 16), 256, 0, stream>>>(c3_Wr, w3rp, 25, 15, 16, 32);

    // ---- layer 1: x(128) -> h1(9), H=3,C=3, pb edges ----
    k_proj_wmma<<<projBlocks, 256, 0, stream>>>(x, 128, w1lp, c1_bl, w1rp, c1_br, xl, xr, 9, Ntiles);
    k_layer_init<<<gblk((long)N * 9), 256, 0, stream>>>(acc, denom, mmax, N, 9, 3);
    k_logit_max<3, 3><<<gblk((long)E1 + N), 256, 0, stream>>>(pb_ei, pb_ea, loop_pb, xl, xr, c1_We, c1_at, mmax, E1, N);
    k_expsum<3, 3><<<gblk((long)E1 + N), 256, 0, stream>>>(pb_ei, pb_ea, loop_pb, xl, xr, c1_We, c1_at, mmax, denom, acc, E1, N);
    k_finalize<<<gblk((long)N * 12), 256, 0, stream>>>(acc, denom, c1_bs, h1p, 12, N, 3, 3);

    // ---- layer 2: h1(9) -> h2(15), H=3,C=5, sp edges ----
    k_proj_wmma<<<projBlocks, 256, 0, stream>>>(h1p, 12, w2lp, c2_bl, w2rp, c2_br, xl, xr, 15, Ntiles);
    k_layer_init<<<gblk((long)N * 15), 256, 0, stream>>>(acc, denom, mmax, N, 15, 3);
    k_logit_max<3, 5><<<gblk((long)E2 + N), 256, 0, stream>>>(sp_ei, sp_ea, loop_sp, xl, xr, c2_We, c2_at, mmax, E2, N);
    k_expsum<3, 5><<<gblk((long)E2 + N), 256, 0, stream>>>(sp_ei, sp_ea, loop_sp, xl, xr, c2_We, c2_at, mmax, denom, acc, E2, N);
    k_finalize<<<gblk((long)N * 16), 256, 0, stream>>>(acc, denom, c2_bs, h2p, 16, N, 3, 5);

    // ---- layer 3: h2(15) -> h3(25), H=5,C=5, sp edges ----
    k_proj_wmma<<<projBlocks, 256, 0, stream>>>(h2p, 16, w3lp, c3_bl, w3rp, c3_br, xl, xr, 25, Ntiles);
    k_layer_init<<<gblk((long)N * 25), 256, 0, stream>>>(acc, denom, mmax, N, 25, 5);
    k_logit_max<5, 5><<<gblk((long)E2 + N), 256, 0, stream>>>(sp_ei, sp_ea, loop_sp, xl, xr, c3_We, c3_at, mmax, E2, N);
    k_expsum<5, 5><<<gblk((long)E2 + N), 256, 0, stream>>>(sp_ei, sp_ea, loop_sp, xl, xr, c3_We, c3_at, mmax, denom, acc, E2, N);
    k_finalize<<<gblk((long)N * 28), 256, 0, stream>>>(acc, denom, c3_bs, h3p, 28, N, 5, 5);

    // ---- readout over if edges ----
    k_edge_mlp<<<gblk(E3), 256, 0, stream>>>(if_ei, if_ea, h3p, e1_W, e1_b, e2_W, e2_b, e3_W, e3_b, out, E3);

    (void)n_in; (void)out_size; (void)ws_size;
}